// AdaptiveEncoderCls_so_91233695301842
// MI455X (gfx1250) — compile-verified
//
#include <hip/hip_runtime.h>
#include <math.h>

// ---------------- constants from the reference ----------------
#define BB        32
#define N0        2048
#define INIT_DIM  32
#define KNN_K     32
#define IN_DIM    3
#define SIGMA_C   0.26f
#define BASELINE_C 0.1f
#define SCALING_C 10.0f
#define EPS_C     1e-6f
#define OUT_COLS  1920       // 2*(64+128+256+512)

typedef float v2f __attribute__((ext_vector_type(2)));
typedef float v8f __attribute__((ext_vector_type(8)));

// adaptive_embedding: one output channel for a 3-vector input
__device__ __forceinline__ float adaptive_pe(const float* xv, int o, int out_dim,
                                             float asig, float blend) {
  int feat_dim = (out_dim + IN_DIM - 1) / IN_DIM;
  int feat_num = feat_dim * IN_DIM;
  int f = (o * (feat_num - 1)) / (out_dim - 1);   // floor(linspace)
  int d = f / feat_dim;
  int j = f - d * feat_dim;
  float fv = -1.0f + 2.0f * (float)(j + 1) / (float)(feat_dim + 1);
  float t = (xv[d] - fv) / (asig + EPS_C);
  return blend * __expf(-0.5f * t * t) + (1.0f - blend) * __cosf(t);
}

// ---------------- small utility kernels ----------------
__global__ void zero_kernel(float* p, int n) {
  for (int i = threadIdx.x; i < n; i += blockDim.x) p[i] = 0.0f;
}

// per-(b,c) sum/sumsq over N points of xyz -> ps/pss[b*3+c]
__global__ void colstd_partial_kernel(const float* __restrict__ x,
                                      float* ps, float* pss, int N) {
  int g = blockIdx.x;            // b*3+c
  int b = g / 3, c = g % 3;
  __shared__ float s1[256], s2[256];
  float a = 0.f, aa = 0.f;
  for (int i = threadIdx.x; i < N; i += blockDim.x) {
    float v = x[((size_t)b * N + i) * 3 + c];
    a += v; aa += v * v;
  }
  s1[threadIdx.x] = a; s2[threadIdx.x] = aa;
  __syncthreads();
  for (int st = 128; st > 0; st >>= 1) {
    if (threadIdx.x < st) { s1[threadIdx.x] += s1[threadIdx.x + st];
                            s2[threadIdx.x] += s2[threadIdx.x + st]; }
    __syncthreads();
  }
  if (threadIdx.x == 0) { ps[g] = s1[0]; pss[g] = s2[0]; }
}

// gstd = mean over M entries of std (ddof=1, count n) -> asig, blend
__global__ void finalize_gstd_kernel(const float* ps, const float* pss,
                                     int M, int n, float* scal) {
  __shared__ float s[128];
  int t = threadIdx.x;
  float v = 0.f;
  if (t < M) {
    float mean = ps[t] / (float)n;
    float var  = (pss[t] - (float)n * mean * mean) / (float)(n - 1);
    v = sqrtf(fmaxf(var, 0.0f));
  }
  s[t] = v; __syncthreads();
  for (int st = 64; st > 0; st >>= 1) { if (t < st) s[t] += s[t + st]; __syncthreads(); }
  if (t == 0) {
    float gstd = s[0] / (float)M;
    scal[0] = SIGMA_C * (1.0f + gstd);                                  // asig
    scal[1] = 1.0f / (1.0f + __expf(-(gstd - BASELINE_C) * SCALING_C)); // blend
  }
}

// initial embedding: feat0[b,n,0:32]
__global__ void embed_init_kernel(const float* __restrict__ xyz, float* feat,
                                  const float* scal, int total) {
  int g = blockIdx.x * blockDim.x + threadIdx.x;   // over B*N
  if (g >= total) return;
  float xv[3] = { xyz[(size_t)g * 3], xyz[(size_t)g * 3 + 1], xyz[(size_t)g * 3 + 2] };
  float asig = scal[0], blend = scal[1];
  float* out = feat + (size_t)g * INIT_DIM;
  for (int o = 0; o < INIT_DIM; ++o) out[o] = adaptive_pe(xv, o, INIT_DIM, asig, blend);
}

// ---------------- furthest point sampling (sequential per batch) ----------------
__global__ void fps_kernel(const float* __restrict__ xyz, int* fps_idx, int N, int S) {
  int b = blockIdx.x, tid = threadIdx.x;
  __shared__ float dist[2048];
  __shared__ float rv[256];
  __shared__ int   ri[256];
  __shared__ int   far;
  __shared__ float c0, c1, c2;
  for (int i = tid; i < N; i += 256) dist[i] = INFINITY;
  if (tid == 0) far = 0;
  __syncthreads();
  for (int j = 0; j < S; ++j) {
    if (tid == 0) {
      fps_idx[b * S + j] = far;
      const float* c = xyz + ((size_t)b * N + far) * 3;
      c0 = c[0]; c1 = c[1]; c2 = c[2];
    }
    __syncthreads();
    float lb = -INFINITY; int li = 0;
    for (int i = tid; i < N; i += 256) {
      const float* p = xyz + ((size_t)b * N + i) * 3;
      float dx = p[0] - c0, dy = p[1] - c1, dz = p[2] - c2;
      float nd = fminf(dist[i], dx * dx + dy * dy + dz * dz);
      dist[i] = nd;
      if (nd > lb) { lb = nd; li = i; }
    }
    rv[tid] = lb; ri[tid] = li;
    __syncthreads();
    for (int st = 128; st > 0; st >>= 1) {
      if (tid < st) {
        if (rv[tid + st] > rv[tid] ||
            (rv[tid + st] == rv[tid] && ri[tid + st] < ri[tid])) {
          rv[tid] = rv[tid + st]; ri[tid] = ri[tid + st];
        }
      }
      __syncthreads();
    }
    if (tid == 0) far = ri[0];
    __syncthreads();
  }
}

// gather sampled xyz and feat rows
__global__ void gather_kernel(const float* __restrict__ xyz, const float* __restrict__ feat,
                              const int* __restrict__ fps_idx,
                              float* xyz_s, float* feat_s, int N, int S, int Cin) {
  int g = blockIdx.x * blockDim.x + threadIdx.x;   // over B*S
  if (g >= BB * S) return;
  int b = g / S, s = g % S;
  int idx = fps_idx[b * S + s];
  const float* p = xyz + ((size_t)b * N + idx) * 3;
  float* q = xyz_s + ((size_t)b * S + s) * 3;
  q[0] = p[0]; q[1] = p[1]; q[2] = p[2];
  const float* f = feat + ((size_t)b * N + idx) * Cin;
  float* fo = feat_s + ((size_t)b * S + s) * Cin;
  for (int c = 0; c < Cin; ++c) fo[c] = f[c];
}

// ---------------- fused WMMA distance + top-K ----------------
// Augmented-vector trick: A row = (qx,qy,qz,1); B col = (-2px,-2py,-2pz,|p|^2)
// => C = |p|^2 - 2<q,p>, ranking-equivalent to squared distance per query row.
#define KNN_CHUNK 512
__global__ void __launch_bounds__(256)
knn_wmma_kernel(const float* __restrict__ xyz, const float* __restrict__ xyz_s,
                int* __restrict__ knn_idx, int N, int S) {
  __shared__ float strip[16 * KNN_CHUNK];     // 32 KB distance strip
  __shared__ float As[64];                    // 16 queries x (x,y,z,1)
  __shared__ float bestd[16][KNN_K];
  __shared__ int   besti[16][KNN_K];
  __shared__ float nbd[16][KNN_K];
  __shared__ int   nbi[16][KNN_K];

  int tiles_s = S >> 4;
  int b    = blockIdx.x / tiles_s;
  int row0 = (blockIdx.x % tiles_s) << 4;
  int tid = threadIdx.x, lane = tid & 31, wave = tid >> 5;
  // wave id is uniform across the wave: pin it to an SGPR so the WMMA tile
  // loop compiles to scalar branches and EXEC stays all-ones (ISA requirement
  // for V_WMMA_*).
  int wave_u = __builtin_amdgcn_readfirstlane(wave);

  if (tid < 16) {
    const float* q = xyz_s + ((size_t)b * S + row0 + tid) * 3;
    As[tid * 4 + 0] = q[0]; As[tid * 4 + 1] = q[1];
    As[tid * 4 + 2] = q[2]; As[tid * 4 + 3] = 1.0f;
  }
  {
    int r0 = wave_u * 2;               // 8 waves x 2 rows = 16 rows
    bestd[r0][lane] = INFINITY;  besti[r0][lane] = 0x7fffffff;
    bestd[r0 + 1][lane] = INFINITY; besti[r0 + 1][lane] = 0x7fffffff;
  }
  __syncthreads();

  int m = lane & 15, hi = lane >> 4;
  // A 16x4 f32 layout: lanes 0-15 hold K=0,1; lanes 16-31 hold K=2,3
  v2f a; a.x = As[m * 4 + hi * 2 + 0]; a.y = As[m * 4 + hi * 2 + 1];

  for (int base = 0; base < N; base += KNN_CHUNK) {
    int chunkN = (N - base) < KNN_CHUNK ? (N - base) : KNN_CHUNK;
    int ntiles = chunkN >> 4;
    // phase 1: WMMA distance tiles into LDS strip (scalar loop, EXEC full)
    for (int t = wave_u; t < ntiles; t += 8) {
      int col = base + t * 16 + m;
      const float* p = xyz + ((size_t)b * N + col) * 3;
      float px = p[0], py = p[1], pz = p[2];
      // prefetch next tile for this wave (uniform predicate -> scalar branch)
      if (t + 8 < ntiles)
        __builtin_prefetch(xyz + ((size_t)b * N + base + (t + 8) * 16 + m) * 3, 0, 1);
      v2f bv;   // B 4x16 layout mirrors A: lanes 0-15 K=0,1; lanes 16-31 K=2,3
      if (hi == 0) { bv.x = -2.0f * px; bv.y = -2.0f * py; }
      else         { bv.x = -2.0f * pz; bv.y = px * px + py * py + pz * pz; }
      v8f c = {};
      c = __builtin_amdgcn_wmma_f32_16x16x4_f32(false, a, false, bv,
                                                (short)0, c, false, false);
#pragma unroll
      for (int v = 0; v < 8; ++v) {
        int row = v + 8 * hi;     // C layout: lanes 0-15 -> M=v, lanes 16-31 -> M=v+8
        strip[row * KNN_CHUNK + t * 16 + m] = c[v];
      }
    }
    __syncthreads();
    // phase 2: per-row selection, merging running best-32 as virtual columns
    for (int rr = 0; rr < 2; ++rr) {
      int row = wave_u * 2 + rr;
      int ext = chunkN + KNN_K;
      for (int k = 0; k < KNN_K; ++k) {
        float best = INFINITY; int bi = 0x7fffffff; int bslot = 0;
        for (int i = lane; i < ext; i += 32) {
          float dv; int iv;
          if (i < chunkN) { dv = strip[row * KNN_CHUNK + i]; iv = base + i; }
          else            { dv = bestd[row][i - chunkN];     iv = besti[row][i - chunkN]; }
          if (dv < best || (dv == best && iv < bi)) { best = dv; bi = iv; bslot = i; }
        }
        for (int off = 16; off > 0; off >>= 1) {   // wave32 argmin reduce
          float ob = __shfl_xor(best, off, 32);
          int   oi = __shfl_xor(bi, off, 32);
          int   os = __shfl_xor(bslot, off, 32);
          if (ob < best || (ob == best && oi < bi)) { best = ob; bi = oi; bslot = os; }
        }
        if (lane == (bslot & 31)) {                // owner lane invalidates
          if (bslot < chunkN) strip[row * KNN_CHUNK + bslot] = INFINITY;
          else                bestd[row][bslot - chunkN] = INFINITY;
        }
        if (lane == 0) { nbd[row][k] = best; nbi[row][k] = bi; }
      }
      bestd[row][lane] = nbd[row][lane];
      besti[row][lane] = nbi[row][lane];
    }
    __syncthreads();
  }
  {
    int r0 = wave_u * 2;
    knn_idx[(((size_t)b * S) + row0 + r0) * KNN_K + lane]     = besti[r0][lane];
    knn_idx[(((size_t)b * S) + row0 + r0 + 1) * KNN_K + lane] = besti[r0 + 1][lane];
  }
}

// ---------------- per-k centered stats (std over (B,S,coords)) ----------------
__global__ void knn_center_stats_kernel(const float* __restrict__ xyz,
                                        const float* __restrict__ xyz_s,
                                        const float* __restrict__ feat,
                                        const float* __restrict__ feat_s,
                                        const int* __restrict__ knn_idx,
                                        float* acc, int N, int S, int Cin) {
  int bs = blockIdx.x; int b = bs / S, s = bs % S;
  int k = threadIdx.x;
  if (k >= KNN_K) return;
  int idx = knn_idx[((size_t)b * S + s) * KNN_K + k];
  const float* p = xyz + ((size_t)b * N + idx) * 3;
  const float* q = xyz_s + ((size_t)b * S + s) * 3;
  float xs = 0.f, xss = 0.f;
  for (int c = 0; c < 3; ++c) { float d = p[c] - q[c]; xs += d; xss += d * d; }
  const float* f  = feat + ((size_t)b * N + idx) * Cin;
  const float* fs = feat_s + ((size_t)b * S + s) * Cin;
  float fsum = 0.f, fss = 0.f;
  for (int c = 0; c < Cin; ++c) { float d = f[c] - fs[c]; fsum += d; fss += d * d; }
  atomicAdd(&acc[k], xs);              atomicAdd(&acc[KNN_K + k], xss);
  atomicAdd(&acc[2 * KNN_K + k], fsum); atomicAdd(&acc[3 * KNN_K + k], fss);
}

__global__ void finalize_knn_std_kernel(const float* acc, float* stdx, float* stdf,
                                        int nx, int nf) {
  int k = threadIdx.x;
  if (k >= KNN_K) return;
  float m = acc[k] / (float)nx;
  float v = (acc[KNN_K + k] - (float)nx * m * m) / (float)(nx - 1);
  stdx[k] = fmaxf(sqrtf(fmaxf(v, 0.f)), 1e-5f);
  float mf = acc[2 * KNN_K + k] / (float)nf;
  float vf = (acc[3 * KNN_K + k] - (float)nf * mf * mf) / (float)(nf - 1);
  stdf[k] = fmaxf(sqrtf(fmaxf(vf, 0.f)), 1e-5f);
}

// per-(b,c) stats of normalized knn offsets over S*K -> for stage gstd
__global__ void knnstd_partial_kernel(const float* __restrict__ xyz,
                                      const float* __restrict__ xyz_s,
                                      const int* __restrict__ knn_idx,
                                      const float* __restrict__ stdx,
                                      float* ps, float* pss, int N, int S) {
  int g = blockIdx.x;          // b*3+c
  int b = g / 3, c = g % 3;
  __shared__ float s1[256], s2[256];
  float a = 0.f, aa = 0.f;
  int total = S * KNN_K;
  for (int i = threadIdx.x; i < total; i += blockDim.x) {
    int s = i / KNN_K, k = i % KNN_K;
    int idx = knn_idx[((size_t)b * S + s) * KNN_K + k];
    float v = (xyz[((size_t)b * N + idx) * 3 + c] -
               xyz_s[((size_t)b * S + s) * 3 + c]) / stdx[k];
    a += v; aa += v * v;
  }
  s1[threadIdx.x] = a; s2[threadIdx.x] = aa;
  __syncthreads();
  for (int st = 128; st > 0; st >>= 1) {
    if (threadIdx.x < st) { s1[threadIdx.x] += s1[threadIdx.x + st];
                            s2[threadIdx.x] += s2[threadIdx.x + st]; }
    __syncthreads();
  }
  if (threadIdx.x == 0) { ps[g] = s1[0]; pss[g] = s2[0]; }
}

// ---------------- stage body: embed + aggregate + pool + gelu ----------------
__global__ void stage_main_kernel(const float* __restrict__ xyz,
                                  const float* __restrict__ xyz_s,
                                  const float* __restrict__ feat,
                                  const float* __restrict__ feat_s,
                                  const int* __restrict__ knn_idx,
                                  const float* __restrict__ stdx,
                                  const float* __restrict__ stdf,
                                  const float* __restrict__ scal,
                                  float* feat_out, int N, int S, int Cin) {
  int Cout = 2 * Cin;
  int bs = blockIdx.x; int b = bs / S, s = bs % S;
  __shared__ float sx[KNN_K][3];
  __shared__ int   sidx[KNN_K];
  __shared__ float sfs[256];
  int tid = threadIdx.x;
  if (tid < KNN_K) {
    int idx = knn_idx[((size_t)b * S + s) * KNN_K + tid];
    sidx[tid] = idx;
    float st = stdx[tid];
    const float* p = xyz + ((size_t)b * N + idx) * 3;
    const float* q = xyz_s + ((size_t)b * S + s) * 3;
    sx[tid][0] = (p[0] - q[0]) / st;
    sx[tid][1] = (p[1] - q[1]) / st;
    sx[tid][2] = (p[2] - q[2]) / st;
  }
  for (int c = tid; c < Cin; c += blockDim.x)
    sfs[c] = feat_s[((size_t)b * S + s) * Cin + c];
  __syncthreads();
  float asig = scal[0], blend = scal[1];
  for (int o = tid; o < Cout; o += blockDim.x) {
    float acc = 0.f, mx = -INFINITY;
    for (int k = 0; k < KNN_K; ++k) {
      float pe = adaptive_pe(sx[k], o, Cout, asig, blend);
      float fc;
      if (o < Cin) fc = (feat[((size_t)b * N + sidx[k]) * Cin + o] - sfs[o]) / stdf[k];
      else         fc = sfs[o - Cin];
      float w = (fc + pe) * pe;
      acc += w; mx = fmaxf(mx, w);
    }
    float pooled = acc / (float)KNN_K + mx;
    float g = 0.5f * pooled * (1.0f + erff(pooled * 0.70710678f));   // exact gelu
    feat_out[((size_t)b * S + s) * Cout + o] = g;
  }
}

// max & mean over S -> output segment
__global__ void stage_out_kernel(const float* __restrict__ feat, float* out,
                                 int S, int Cout, int seg) {
  int b = blockIdx.x;
  for (int o = threadIdx.x; o < Cout; o += blockDim.x) {
    float mx = -INFINITY, sm = 0.f;
    for (int s = 0; s < S; ++s) {
      float v = feat[((size_t)b * S + s) * Cout + o];
      mx = fmaxf(mx, v); sm += v;
    }
    out[(size_t)b * OUT_COLS + seg + o] = mx;
    out[(size_t)b * OUT_COLS + seg + Cout + o] = sm / (float)S;
  }
}

// ---------------- host orchestration ----------------
extern "C" void kernel_launch(void* const* d_in, const int* in_sizes, int n_in,
                              void* d_out, int out_size, void* d_ws, size_t ws_size,
                              hipStream_t stream) {
  (void)in_sizes; (void)n_in; (void)out_size; (void)ws_size;
  const float* xyz_in = (const float*)d_in[0];
  float* out = (float*)d_out;
  float* ws  = (float*)d_ws;

  // workspace layout (floats); total ~27 MB — fits easily in MI455X L2
  float* featA  = ws;                      // 2,097,152
  float* featB  = featA + 2097152;         // 2,097,152
  float* xyzA   = featB + 2097152;         //   196,608
  float* xyzB   = xyzA + 196608;           //   196,608
  float* feat_s = xyzB + 196608;           // 1,048,576
  float* stats  = feat_s + 1048576;
  float* acc4k  = stats;                   // 128
  float* stdx   = stats + 128;             // 32
  float* stdf   = stats + 160;             // 32
  float* ps     = stats + 192;             // 96
  float* pss    = stats + 288;             // 96
  float* scal   = stats + 384;             // 2 (padded to 400)
  int*   iptr    = (int*)(stats + 400);
  int*   fps_idx = iptr;                   // 32,768
  int*   knn_idx = iptr + 32768;           // 1,048,576

  // initial embedding of raw xyz (gstd over per-(b,c) stds, count N0)
  zero_kernel<<<1, 256, 0, stream>>>(ps, 192);
  colstd_partial_kernel<<<BB * 3, 256, 0, stream>>>(xyz_in, ps, pss, N0);
  finalize_gstd_kernel<<<1, 128, 0, stream>>>(ps, pss, 96, N0, scal);
  embed_init_kernel<<<(BB * N0 + 255) / 256, 256, 0, stream>>>(xyz_in, featA, scal, BB * N0);

  const float* cur_xyz = xyz_in;
  float* cur_feat = featA;
  float* nxt_feat = featB;
  int N = N0, Cin = INIT_DIM, seg = 0;

  for (int st = 0; st < 4; ++st) {
    int S = N / 2, Cout = 2 * Cin;
    float* xyz_s = (st & 1) ? xyzB : xyzA;

    fps_kernel<<<BB, 256, 0, stream>>>(cur_xyz, fps_idx, N, S);
    gather_kernel<<<(BB * S + 63) / 64, 64, 0, stream>>>(cur_xyz, cur_feat, fps_idx,
                                                         xyz_s, feat_s, N, S, Cin);
    knn_wmma_kernel<<<BB * (S / 16), 256, 0, stream>>>(cur_xyz, xyz_s, knn_idx, N, S);

    zero_kernel<<<1, 256, 0, stream>>>(acc4k, 128);
    knn_center_stats_kernel<<<BB * S, 32, 0, stream>>>(cur_xyz, xyz_s, cur_feat, feat_s,
                                                       knn_idx, acc4k, N, S, Cin);
    finalize_knn_std_kernel<<<1, 32, 0, stream>>>(acc4k, stdx, stdf, BB * S * 3, BB * S * Cin);

    zero_kernel<<<1, 256, 0, stream>>>(ps, 192);
    knnstd_partial_kernel<<<BB * 3, 256, 0, stream>>>(cur_xyz, xyz_s, knn_idx, stdx,
                                                      ps, pss, N, S);
    finalize_gstd_kernel<<<1, 128, 0, stream>>>(ps, pss, 96, S * KNN_K, scal);

    stage_main_kernel<<<BB * S, 128, 0, stream>>>(cur_xyz, xyz_s, cur_feat, feat_s,
                                                  knn_idx, stdx, stdf, scal,
                                                  nxt_feat, N, S, Cin);
    stage_out_kernel<<<BB, 256, 0, stream>>>(nxt_feat, out, S, Cout, seg);

    seg += 2 * Cout;
    cur_xyz = xyz_s;
    float* t = cur_feat; cur_feat = nxt_feat; nxt_feat = t;
    N = S; Cin = Cout;
  }
}